// HVAE_42159398978260
// MI455X (gfx1250) — compile-verified
//
#include <hip/hip_runtime.h>
#include <math.h>

// ---------------------------------------------------------------------------
// Problem constants (fixed by the reference)
// ---------------------------------------------------------------------------
#define N_NODES 8192
#define DLAT    64
static constexpr float LOG2PI_F = 1.8378770664093453f;   // log(2*pi)
static constexpr float LVC_F    = -1.3862943611198906f;  // log(0.25)

typedef __attribute__((ext_vector_type(2))) float v2f;
typedef __attribute__((ext_vector_type(8))) float v8f;

// ---------------------------------------------------------------------------
// Exact wave32 sum via V_WMMA_F32_16X16X4_F32.
// A = ones(16x4), B holds the 32 lane values (b.y = 0 pads to 64 B-slots).
// D[m][n] = sum_k B[k][n]  -> every row of D is the column-sum vector.
// Sum of c[0] over all 32 lanes = row0-sum + row8-sum = exactly 2 * wave total.
// The xor-tree first adds identical values (exact doubling), *0.5f is exact.
// ---------------------------------------------------------------------------
__device__ __forceinline__ float wave_sum(float v) {
#if __has_builtin(__builtin_amdgcn_wmma_f32_16x16x4_f32)
  v2f a; a[0] = 1.0f; a[1] = 1.0f;
  v2f b; b[0] = v;    b[1] = 0.0f;
  v8f c = {};
  c = __builtin_amdgcn_wmma_f32_16x16x4_f32(false, a, false, b,
                                            (short)0, c, false, false);
  float s = c[0];
#pragma unroll
  for (int off = 16; off > 0; off >>= 1) s += __shfl_xor(s, off, 32);
  return s * 0.5f;
#else
  float s = v;
#pragma unroll
  for (int off = 16; off > 0; off >>= 1) s += __shfl_xor(s, off, 32);
  return s;
#endif
}

// Block sum; result valid in thread 0 only. lds must hold blockDim.x/32 floats.
__device__ __forceinline__ float block_sum(float v, float* lds) {
  const int lane = threadIdx.x & 31;
  const int wid  = threadIdx.x >> 5;
  float w = wave_sum(v);               // all 32 lanes active -> EXEC all ones
  if (lane == 0) lds[wid] = w;
  __syncthreads();
  float r = 0.0f;
  if (threadIdx.x == 0) {
    const int nw = (int)(blockDim.x >> 5);
    for (int i = 0; i < nw; ++i) r += lds[i];
  }
  return r;
}

// ---------------------------------------------------------------------------
// Kernel 0: zero the adjacency bitmask (N*N bits = 2M words = 8 MB)
// ---------------------------------------------------------------------------
__global__ void zero_mask_kernel(uint4* __restrict__ p, int n4) {
  const uint4 z = {0u, 0u, 0u, 0u};
  for (int i = blockIdx.x * blockDim.x + threadIdx.x; i < n4;
       i += gridDim.x * blockDim.x)
    p[i] = z;
}

// ---------------------------------------------------------------------------
// Kernel 1: scatter symmetric edges as bits (idempotent OR == ".set(1.0)")
// ---------------------------------------------------------------------------
__global__ void scatter_edges_kernel(const int* __restrict__ ei, int E,
                                     unsigned* __restrict__ mask) {
  int e = blockIdx.x * blockDim.x + threadIdx.x;
  if (e >= E) return;
  unsigned i = (unsigned)ei[e];
  unsigned j = (unsigned)ei[E + e];
  atomicOr(&mask[i * (N_NODES / 32) + (j >> 5)], 1u << (j & 31u));
  atomicOr(&mask[j * (N_NODES / 32) + (i >> 5)], 1u << (i & 31u));
}

// ---------------------------------------------------------------------------
// Kernel 2: per-node KL term.  kld = -0.5*(1 + zl - lvc) + 2*((zm-pm)^2 + e^zl)
// ---------------------------------------------------------------------------
__global__ void kl_kernel(const float* __restrict__ zm,
                          const float* __restrict__ zl,
                          const float* __restrict__ pm,
                          float* __restrict__ partial, float scale) {
  __shared__ float lds[8];
  const int total = N_NODES * DLAT;
  float acc = 0.0f;
  for (int idx = blockIdx.x * blockDim.x + threadIdx.x; idx < total;
       idx += gridDim.x * blockDim.x) {
    float m = zm[idx];
    float l = zl[idx];
    float p = pm[idx & (DLAT - 1)];
    float dm = m - p;
    acc += -0.5f * (1.0f + l - LVC_F) + 2.0f * (dm * dm + __expf(l));
  }
  float r = block_sum(acc, lds);
  if (threadIdx.x == 0) partial[blockIdx.x] = r * scale;
}

// ---------------------------------------------------------------------------
// Kernel 3: tiny D=64 vector terms.
//   log_p    = sum_v mean_d( -0.5*(log2pi + m^2) )
//   extra_kl = sum_pairs mean_d( 2*(m-q)^2 )
// One block of 64 threads.
// ---------------------------------------------------------------------------
__global__ void vec_terms_kernel(const float* __restrict__ m0, const float* __restrict__ m1,
                                 const float* __restrict__ m2, const float* __restrict__ m3,
                                 const float* __restrict__ m4, const float* __restrict__ m5,
                                 const float* __restrict__ q0, const float* __restrict__ q1,
                                 const float* __restrict__ q2, const float* __restrict__ q3,
                                 const float* __restrict__ q4, const float* __restrict__ q5,
                                 float* __restrict__ partial) {
  __shared__ float lds[2];
  const int d = threadIdx.x;  // 64 threads
  const float* ms[6] = {m0, m1, m2, m3, m4, m5};
  const float* qs[6] = {q0, q1, q2, q3, q4, q5};
  float acc = 0.0f;
#pragma unroll
  for (int v = 0; v < 6; ++v) {
    float m = ms[v][d];
    float q = qs[v][d];
    acc += -0.5f * (LOG2PI_F + m * m);   // log p(mu)
    float dm = m - q;
    acc += 2.0f * dm * dm;               // extra KL
  }
  float r = block_sum(acc, lds);
  if (threadIdx.x == 0) partial[0] = r * (1.0f / (float)DLAT);
}

// ---------------------------------------------------------------------------
// Kernel 4: BCE-with-logits over 8192x8192 against bitmask adjacency.
//   max(x,0) - x*a + log(1 + exp(-|x|))   (1+e well-conditioned, e in (0,1])
// float4 loads of x; 4 adjacent bits live in one mask word (aligned).
// ---------------------------------------------------------------------------
__global__ void bce_kernel(const float4* __restrict__ x4,
                           const unsigned* __restrict__ mask,
                           float* __restrict__ partial, float scale) {
  __shared__ float lds[8];
  const int total4 = (N_NODES * N_NODES) / 4;  // 16,777,216
  float acc = 0.0f;
  for (int i = blockIdx.x * blockDim.x + threadIdx.x; i < total4;
       i += gridDim.x * blockDim.x) {
    float4 v = x4[i];
    unsigned w = mask[i >> 3];      // 8 float4 per 32-bit mask word
    int b0 = (i & 7) * 4;           // starting bit of this quad
    float xs[4] = {v.x, v.y, v.z, v.w};
#pragma unroll
    for (int j = 0; j < 4; ++j) {
      float xv = xs[j];
      float a = (float)((w >> (b0 + j)) & 1u);
      acc += fmaxf(xv, 0.0f) - xv * a +
             __logf(1.0f + __expf(-fabsf(xv)));
    }
  }
  float r = block_sum(acc, lds);
  if (threadIdx.x == 0) partial[blockIdx.x] = r * scale;
}

// ---------------------------------------------------------------------------
// Kernel 5: deterministic fp64 reduction of all partials -> scalar out
// ---------------------------------------------------------------------------
__global__ void final_reduce_kernel(const float* __restrict__ partials, int n,
                                    float* __restrict__ out) {
  __shared__ double lds[256];
  double acc = 0.0;
  for (int i = threadIdx.x; i < n; i += blockDim.x)
    acc += (double)partials[i];
  lds[threadIdx.x] = acc;
  __syncthreads();
  for (int s = 128; s > 0; s >>= 1) {
    if ((int)threadIdx.x < s) lds[threadIdx.x] += lds[threadIdx.x + s];
    __syncthreads();
  }
  if (threadIdx.x == 0) out[0] = (float)lds[0];
}

// ---------------------------------------------------------------------------
// Launch
// ---------------------------------------------------------------------------
extern "C" void kernel_launch(void* const* d_in, const int* in_sizes, int n_in,
                              void* d_out, int out_size, void* d_ws, size_t ws_size,
                              hipStream_t stream) {
  (void)n_in; (void)out_size; (void)ws_size;

  // Input order (see setup_inputs): 0..11 z arrays, 12..17 Alpha/Beta_mu*,
  // 18..23 mu_Alpha/Beta*, 24 edge_logits, 25 edge_index, 26 num_nodes.
  const float* zin[12];
  for (int i = 0; i < 12; ++i) zin[i] = (const float*)d_in[i];
  const float* Amu[6];
  for (int i = 0; i < 6; ++i) Amu[i] = (const float*)d_in[12 + i];
  const float* muA[6];
  for (int i = 0; i < 6; ++i) muA[i] = (const float*)d_in[18 + i];
  const float* edge_logits = (const float*)d_in[24];
  const int*   edge_index  = (const int*)d_in[25];
  const int    E = in_sizes[25] / 2;

  // Workspace: [ 8 MB bitmask | partials ]
  unsigned* mask = (unsigned*)d_ws;
  const int maskWords = (N_NODES / 32) * N_NODES;            // 2,097,152
  float* partials = (float*)((char*)d_ws + (size_t)maskWords * sizeof(unsigned));

  const int bceBlocks = 4096;
  const int klBlocks  = 256;
  float* bcePart = partials;                       // [0, 4096)
  float* klPart  = partials + bceBlocks;           // 6 * 256
  float* vecPart = partials + bceBlocks + 6 * klBlocks;  // 1
  const int nPart = bceBlocks + 6 * klBlocks + 1;  // 5633

  // 0) zero bitmask
  zero_mask_kernel<<<512, 256, 0, stream>>>((uint4*)mask, maskWords / 4);

  // 1) scatter symmetric edges
  scatter_edges_kernel<<<(E + 255) / 256, 256, 0, stream>>>(edge_index, E, mask);

  // 2) six per-node KL terms: (z_mu_e,z_logvar_e,mu_Alpha), (z_mu_n,..,mu_Beta), ...
  const float klScale = 1.0f / (float)(N_NODES * DLAT);
  kl_kernel<<<klBlocks, 256, 0, stream>>>(zin[2],  zin[3],  muA[0], klPart + 0 * klBlocks, klScale);
  kl_kernel<<<klBlocks, 256, 0, stream>>>(zin[0],  zin[1],  muA[1], klPart + 1 * klBlocks, klScale);
  kl_kernel<<<klBlocks, 256, 0, stream>>>(zin[6],  zin[7],  muA[2], klPart + 2 * klBlocks, klScale);
  kl_kernel<<<klBlocks, 256, 0, stream>>>(zin[4],  zin[5],  muA[3], klPart + 3 * klBlocks, klScale);
  kl_kernel<<<klBlocks, 256, 0, stream>>>(zin[10], zin[11], muA[4], klPart + 4 * klBlocks, klScale);
  kl_kernel<<<klBlocks, 256, 0, stream>>>(zin[8],  zin[9],  muA[5], klPart + 5 * klBlocks, klScale);

  // 3) tiny vector terms (log_p + extra_kl), pairs (mu_Alpha, Alpha_mu) etc.
  vec_terms_kernel<<<1, 64, 0, stream>>>(muA[0], muA[1], muA[2], muA[3], muA[4], muA[5],
                                         Amu[0], Amu[1], Amu[2], Amu[3], Amu[4], Amu[5],
                                         vecPart);

  // 4) BCE over 8192x8192 (after mask is built; same-stream ordering)
  const float bceScale = 1.0f / ((float)N_NODES * (float)N_NODES);
  bce_kernel<<<bceBlocks, 256, 0, stream>>>((const float4*)edge_logits, mask,
                                            bcePart, bceScale);

  // 5) deterministic final reduction
  final_reduce_kernel<<<1, 256, 0, stream>>>(partials, nPart, (float*)d_out);
}